// KNNGraph_90503550861392
// MI455X (gfx1250) — compile-verified
//
#include <hip/hip_runtime.h>

typedef float v2f __attribute__((ext_vector_type(2)));
typedef float v8f __attribute__((ext_vector_type(8)));

#define BATCH 4
#define CH    64
#define NPTS  8192
#define KNN   16
#define EPSF  1e-12f

// ---------------------------------------------------------------------------
// Phase 1: fused pairwise-distance (WMMA f32 16x16x4) + per-row top-16 select.
// One wave (32 lanes) owns a tile of 16 query points; loops over 512 candidate
// tiles of 16. A = -2 * query coords (16x4, z-dim padded), B = candidate
// coords (4x16). D tile is transposed through LDS; each row is scanned by two
// owner lanes that keep a sorted top-16 (dist, idx) list in registers.
// ---------------------------------------------------------------------------
__global__ __launch_bounds__(32)
void knn_topk_wmma(const float* __restrict__ pts, int* __restrict__ idx_out) {
  __shared__ float tile[16][17];   // padded: stride 17 -> conflict-free
  __shared__ int   itile[16][17];
  __shared__ float csq_sh[16];

  const int  b     = blockIdx.y;
  const int  qbase = blockIdx.x << 4;
  const int  lane  = threadIdx.x;
  const int  l15   = lane & 15;
  const bool hi    = lane >= 16;

  const float* __restrict__ px = pts + (size_t)b * 3 * NPTS;
  const float* __restrict__ py = px + NPTS;
  const float* __restrict__ pz = px + 2 * NPTS;

  // This lane's query (row M = l15 of the A matrix).
  const int   q   = qbase + l15;
  const float qx  = px[q], qy = py[q], qz = pz[q];
  const float qsq = qx * qx + qy * qy + qz * qz;

  // A (16x4 f32): lanes 0-15 hold K=0,1 ; lanes 16-31 hold K=2,3 (K=3 pad=0).
  v2f amat;
  amat.x = -2.0f * (hi ? qz : qx);
  amat.y = hi ? 0.0f : (-2.0f * qy);

  // Row ownership for the top-k scan: lane owns row r, columns [j0, j0+8).
  const int r     = l15;
  const int j0    = hi ? 8 : 0;
  const int own_q = q;

  float bd[KNN];
  int   bi[KNN];
#pragma unroll
  for (int k = 0; k < KNN; ++k) { bd[k] = 3.402823466e38f; bi[k] = 0; }

  for (int mbase = 0; mbase < NPTS; mbase += 16) {
    const int   m  = mbase + l15;
    const float cx = px[m], cy = py[m], cz = pz[m];
    if (mbase + 16 < NPTS) {                     // global_prefetch_b8 next tile
      __builtin_prefetch(px + m + 16, 0, 0);
      __builtin_prefetch(py + m + 16, 0, 0);
      __builtin_prefetch(pz + m + 16, 0, 0);
    }

    // B (4x16 f32): lanes 0-15 hold K=0,1 ; lanes 16-31 hold K=2,3 (pad=0).
    v2f bmat;
    bmat.x = hi ? cz : cx;
    bmat.y = hi ? 0.0f : cy;

    v8f acc = {};
    acc = __builtin_amdgcn_wmma_f32_16x16x4_f32(false, amat, false, bmat,
                                                (short)0, acc, false, false);

    if (!hi) csq_sh[l15] = cx * cx + cy * cy + cz * cz;

    // D layout: lane holds column N = l15 for rows M = v + 8*hi. Transpose via LDS.
#pragma unroll
    for (int v = 0; v < 8; ++v)
      tile[v + (hi ? 8 : 0)][l15] = acc[v];
    __syncthreads();

#pragma unroll
    for (int j = 0; j < 8; ++j) {
      const int col  = j0 + j;
      const int midx = mbase + col;
      float d = tile[r][col] + qsq + csq_sh[col];
      d = fmaxf(d, EPSF);
      if (midx != own_q && d < bd[KNN - 1]) {
        float cd = d; int ci = midx;
#pragma unroll
        for (int t = 0; t < KNN; ++t) {          // stable bubble insert
          const bool  sw = cd < bd[t];
          const float od = bd[t];
          const int   oi = bi[t];
          bd[t] = sw ? cd : od;
          bi[t] = sw ? ci : oi;
          cd    = sw ? od : cd;
          ci    = sw ? oi : ci;
        }
      }
    }
    __syncthreads();
  }

  // Merge hi-half lists into lo-half owners via LDS.
  if (hi) {
#pragma unroll
    for (int k = 0; k < KNN; ++k) { tile[r][k] = bd[k]; itile[r][k] = bi[k]; }
  }
  __syncthreads();
  if (!hi) {
#pragma unroll
    for (int k = 0; k < KNN; ++k) {
      float cd = tile[r][k]; int ci = itile[r][k];
      if (cd < bd[KNN - 1]) {
#pragma unroll
        for (int t = 0; t < KNN; ++t) {
          const bool  sw = cd < bd[t];
          const float od = bd[t];
          const int   oi = bi[t];
          bd[t] = sw ? cd : od;
          bi[t] = sw ? ci : oi;
          cd    = sw ? od : cd;
          ci    = sw ? oi : ci;
        }
      }
    }
    int* __restrict__ row = idx_out + ((size_t)b * NPTS + own_q) * KNN;
#pragma unroll
    for (int k = 0; k < KNN; ++k) row[k] = bi[k];
  }
}

// ---------------------------------------------------------------------------
// Phase 2: gather + concat([center, neighbor - center]). Pure store-bound:
// each block writes 4 KB contiguous; neighbor reads scatter only inside one
// 32 KB L2-resident channel row.
// out[b][c2][n][k], c2 < 64 -> center, c2 >= 64 -> f[idx] - center.
// ---------------------------------------------------------------------------
__global__ __launch_bounds__(256)
void knn_gather(const float* __restrict__ feats, const int* __restrict__ idx,
                float* __restrict__ out) {
  const int b   = blockIdx.z;
  const int c2  = blockIdx.y;
  const int tid = threadIdx.x;
  const int n   = blockIdx.x * 64 + (tid >> 2);
  const int k4  = (tid & 3) << 2;

  const float* __restrict__ f = feats + ((size_t)b * CH + (c2 & (CH - 1))) * NPTS;
  const float ctr = f[n];
  float4 v;
  if (c2 < CH) {
    v = make_float4(ctr, ctr, ctr, ctr);
  } else {
    const int4 m = *reinterpret_cast<const int4*>(idx + ((size_t)b * NPTS + n) * KNN + k4);
    v = make_float4(f[m.x] - ctr, f[m.y] - ctr, f[m.z] - ctr, f[m.w] - ctr);
  }
  *reinterpret_cast<float4*>(out + (((size_t)b * 2 * CH + c2) * NPTS + n) * KNN + k4) = v;
}

extern "C" void kernel_launch(void* const* d_in, const int* in_sizes, int n_in,
                              void* d_out, int out_size, void* d_ws, size_t ws_size,
                              hipStream_t stream) {
  (void)in_sizes; (void)n_in; (void)out_size; (void)ws_size;
  const float* pts   = (const float*)d_in[0];   // (4, 3, 8192) f32
  const float* feats = (const float*)d_in[1];   // (4, 64, 8192) f32
  float* out = (float*)d_out;                   // (4, 128, 8192, 16) f32
  int*   idx = (int*)d_ws;                      // 4*8192*16 ints = 2 MiB scratch

  dim3 gridA(NPTS / 16, BATCH);
  knn_topk_wmma<<<gridA, 32, 0, stream>>>(pts, idx);

  dim3 gridB(NPTS / 64, 2 * CH, BATCH);
  knn_gather<<<gridB, 256, 0, stream>>>(feats, idx, out);
}